// VectorQuantizer_21182778704482
// MI455X (gfx1250) — compile-verified
//
#include <hip/hip_runtime.h>
#include <hip/hip_bf16.h>

typedef __attribute__((ext_vector_type(2))) float v2f;
typedef __attribute__((ext_vector_type(8))) float v8f;

// Problem constants (match reference)
#define NUM_E   1024
#define EDIM    64
#define BB      32
#define HW      4096            // H*W
#define CHW     (EDIM * HW)     // 262144
#define NPTS    (BB * HW)       // 131072

// Output layout (flat concat of (z_q_st, vq_loss, indices))
#define ZQ_OFF   0
#define LOSS_OFF (BB * EDIM * HW)   // 8388608
#define IDX_OFF  (LOSS_OFF + 1)

// Workspace layout (floats)
#define WS_CNORM    0       // 1024 floats: ||e_k||^2
#define WS_PARTIAL  1024    // 1024 floats: per-block loss partials

// Tiling
#define STAGE_CODES      128
#define NSTAGES          (NUM_E / STAGE_CODES)      // 8
#define TILES_PER_STAGE  (STAGE_CODES / 16)         // 8
#define LDS_STRIDE       68     // stride%4==0 -> aligned float4 staging; banks (4n+c)%64 conflict-free

#define WMMA_F32X4 __builtin_amdgcn_wmma_f32_16x16x4_f32

// ---------------------------------------------------------------------------
// Kernel 1: codebook squared norms -> d_ws[WS_CNORM + k]
// ---------------------------------------------------------------------------
__global__ __launch_bounds__(256) void vq_norms_kernel(
    const float* __restrict__ cb, float* __restrict__ ws)
{
    int k = blockIdx.x * blockDim.x + threadIdx.x;
    if (k < NUM_E) {
        const float4* row = (const float4*)(cb + k * EDIM);
        float s = 0.f;
        #pragma unroll
        for (int i = 0; i < EDIM / 4; ++i) {
            float4 v = row[i];
            s = fmaf(v.x, v.x, s); s = fmaf(v.y, v.y, s);
            s = fmaf(v.z, v.z, s); s = fmaf(v.w, v.w, s);
        }
        ws[WS_CNORM + k] = s;
    }
}

// ---------------------------------------------------------------------------
// Kernel 2: main VQ kernel. 256 threads = 8 waves; wave owns 16 rows of z_flat.
// Each iteration processes TWO 16-code tiles with independent accumulators so
// the two WMMA chains interleave (A operand shared between the pair).
// ---------------------------------------------------------------------------
__global__ __launch_bounds__(256) void vq_main_kernel(
    const float* __restrict__ z, const float* __restrict__ cb,
    float* __restrict__ ws, float* __restrict__ out)
{
    __shared__ float cbs[STAGE_CODES * LDS_STRIDE];  // 34816 B
    __shared__ float cns[STAGE_CODES];               // staged ||e||^2
    __shared__ float wsum[8];

    const int t    = threadIdx.x;
    const int wave = t >> 5;
    const int lane = t & 31;
    const int l16  = lane & 15;
    const int hi   = lane >> 4;

    const int rowBase = blockIdx.x * 128 + wave * 16;
    const int myrow   = rowBase + l16;          // row this lane's A data belongs to
    const int b       = myrow >> 12;            // / HW
    const int hw      = myrow & (HW - 1);
    const int zbase   = b * CHW + hw;           // + c*HW indexes (B,C,H,W)

    // --- Load A tile: a[2*kk+j] = z_flat[myrow][4*kk + 2*hi + j] (reused twice) ---
    float a[32];
    #pragma unroll
    for (int kk = 0; kk < 16; ++kk) {
        int c = 4 * kk + 2 * hi;
        a[2 * kk]     = z[zbase + c * HW];
        a[2 * kk + 1] = z[zbase + (c + 1) * HW];
    }

    // Running argmin of (||e||^2 - 2 z.e) per C/D row slot
    float bestv[8];
    int   bestn[8];
    #pragma unroll
    for (int i = 0; i < 8; ++i) { bestv[i] = 3.4e38f; bestn[i] = 0; }

    for (int stage = 0; stage < NSTAGES; ++stage) {
        __syncthreads();
        // Cooperative stage of 128 codebook rows + norms into LDS (stride 68)
        {
            const int n  = t >> 1;            // 0..127
            const int c0 = (t & 1) * 32;
            const float4* src = (const float4*)(cb + (stage * STAGE_CODES + n) * EDIM + c0);
            float4*       dst = (float4*)(cbs + n * LDS_STRIDE + c0);
            #pragma unroll
            for (int q = 0; q < 8; ++q) dst[q] = src[q];
            if (t < STAGE_CODES) cns[t] = ws[WS_CNORM + stage * STAGE_CODES + t];
        }
        __syncthreads();

        for (int tt = 0; tt < TILES_PER_STAGE; tt += 2) {
            // Preload both B operand tiles into registers (DS loads pipeline,
            // one wait before the WMMA chains instead of one per pair).
            const v2f* bp0 = (const v2f*)(cbs + ( tt      * 16 + l16) * LDS_STRIDE + 2 * hi);
            const v2f* bp1 = (const v2f*)(cbs + ((tt + 1) * 16 + l16) * LDS_STRIDE + 2 * hi);
            v2f b0[16], b1[16];
            #pragma unroll
            for (int kk = 0; kk < 16; ++kk) { b0[kk] = bp0[2 * kk]; b1[kk] = bp1[2 * kk]; }
            const float cn0 = cns[tt * 16 + l16];
            const float cn1 = cns[tt * 16 + 16 + l16];

            v8f acc0 = {}, acc1 = {};
            #pragma unroll
            for (int kk = 0; kk < 16; ++kk) {
                v2f av; av.x = a[2 * kk]; av.y = a[2 * kk + 1];
                acc0 = WMMA_F32X4(false, av, false, b0[kk], (short)0, acc0, false, false);
                acc1 = WMMA_F32X4(false, av, false, b1[kk], (short)0, acc1, false, false);
            }

            const int nid0 = stage * STAGE_CODES + tt * 16 + l16;
            const int nid1 = nid0 + 16;
            #pragma unroll
            for (int i = 0; i < 8; ++i) {
                float d0 = fmaf(-2.f, acc0[i], cn0);   // == dist - ||z||^2 (row-const)
                if (d0 < bestv[i]) { bestv[i] = d0; bestn[i] = nid0; }
                float d1 = fmaf(-2.f, acc1[i], cn1);
                if (d1 < bestv[i]) { bestv[i] = d1; bestn[i] = nid1; }
            }
        }
    }

    // --- Cross-lane argmin within each 16-lane half (codes differ per lane) ---
    #pragma unroll
    for (int m = 1; m < 16; m <<= 1) {
        #pragma unroll
        for (int i = 0; i < 8; ++i) {
            float ov = __shfl_xor(bestv[i], m, 32);
            int   on = __shfl_xor(bestn[i], m, 32);
            if (ov < bestv[i] || (ov == bestv[i] && on < bestn[i])) {
                bestv[i] = ov; bestn[i] = on;
            }
        }
    }

    // Redistribute: lane needs the winning index of row l16.
    // Rows 0..7 live in half 0 (VGPR i = row), rows 8..15 in half 1.
    int myn = 0;
    #pragma unroll
    for (int r = 0; r < 16; ++r) {
        int v = __shfl(bestn[r & 7], (r < 8) ? 0 : 16, 32);
        if (l16 == r) myn = v;
    }

    // Indices output (flat n order == (b, h*w)); written as float per output dtype
    if (hi == 0) out[IDX_OFF + rowBase + l16] = (float)myn;

    // z_q scatter back to (B,C,H,W) + loss partial, reusing z from registers
    const float* crow = cb + myn * EDIM;
    float lsum = 0.f;
    #pragma unroll
    for (int kk = 0; kk < 16; ++kk) {
        int c = 4 * kk + 2 * hi;
        float q0 = crow[c], q1 = crow[c + 1];
        out[ZQ_OFF + zbase + c * HW]       = q0;
        out[ZQ_OFF + zbase + (c + 1) * HW] = q1;
        float d0 = q0 - a[2 * kk], d1 = q1 - a[2 * kk + 1];
        lsum = fmaf(d0, d0, lsum);
        lsum = fmaf(d1, d1, lsum);
    }

    // Deterministic loss reduction: wave shuffle -> block LDS -> per-block partial
    #pragma unroll
    for (int m = 1; m < 32; m <<= 1) lsum += __shfl_xor(lsum, m, 32);
    __syncthreads();
    if (lane == 0) wsum[wave] = lsum;
    __syncthreads();
    if (t == 0) {
        float p = 0.f;
        #pragma unroll
        for (int w = 0; w < 8; ++w) p += wsum[w];
        ws[WS_PARTIAL + blockIdx.x] = p;
    }
}

// ---------------------------------------------------------------------------
// Kernel 3: fixed-order loss finalize (bit-deterministic across replays)
// ---------------------------------------------------------------------------
__global__ __launch_bounds__(256) void vq_loss_finalize(
    const float* __restrict__ ws, float* __restrict__ out)
{
    __shared__ float s[256];
    int t = threadIdx.x;
    float v = 0.f;
    #pragma unroll
    for (int i = 0; i < 4; ++i) v += ws[WS_PARTIAL + t * 4 + i];
    s[t] = v;
    __syncthreads();
    for (int off = 128; off > 0; off >>= 1) {
        if (t < off) s[t] += s[t + off];
        __syncthreads();
    }
    if (t == 0) out[LOSS_OFF] = s[0] * (1.25f / (float)(BB * EDIM * HW));
}

// ---------------------------------------------------------------------------
extern "C" void kernel_launch(void* const* d_in, const int* in_sizes, int n_in,
                              void* d_out, int out_size, void* d_ws, size_t ws_size,
                              hipStream_t stream)
{
    const float* z  = (const float*)d_in[0];   // (32,64,64,64) fp32
    const float* cb = (const float*)d_in[1];   // (1024,64) fp32
    float* out = (float*)d_out;
    float* ws  = (float*)d_ws;                 // needs 8 KB

    vq_norms_kernel<<<NUM_E / 256, 256, 0, stream>>>(cb, ws);
    vq_main_kernel<<<NPTS / 128, 256, 0, stream>>>(z, cb, ws, out);
    vq_loss_finalize<<<1, 256, 0, stream>>>(ws, out);
}